// DenseGCN_85435489452329
// MI455X (gfx1250) — compile-verified
//
#include <hip/hip_runtime.h>
#include <math.h>

// Problem constants (match reference)
#define NN 65536
#define NE 1048576
#define FIN 128
#define HD 64
#define OD 32
#define NL 3
#define LN_EPS 1e-5f

typedef float v2f __attribute__((ext_vector_type(2)));
typedef float v4f __attribute__((ext_vector_type(4)));
typedef float v8f __attribute__((ext_vector_type(8)));

__device__ __forceinline__ float wave_sum32(float v) {
#pragma unroll
  for (int off = 16; off > 0; off >>= 1) v += __shfl_xor(v, off, 32);
  return v;
}

__device__ __forceinline__ float elu1(float x) {
  return x > 0.f ? x : (__expf(x) - 1.f);
}

// D = A(16x4 f32) * B(4x16 f32) + C(16x16 f32), full-precision CDNA5 WMMA
__device__ __forceinline__ v8f wmma4(v2f a, v2f b, v8f c) {
  return __builtin_amdgcn_wmma_f32_16x16x4_f32(false, a, false, b, (short)0, c,
                                               false, false);
}

// Async 16B global->LDS copy (GLOBAL_LOAD_ASYNC_TO_LDS_B128, ASYNCcnt).
// GV mode: vdst = LDS byte address (low 32 bits of generic LDS ptr),
// vaddr = 64-bit global address, saddr = off (NULL).
__device__ __forceinline__ void async_copy16(float* ldst, const float* gsrc) {
  asm volatile("global_load_async_to_lds_b128 %0, %1, off" ::"v"(
                   (unsigned)(uintptr_t)ldst),
               "v"((unsigned long long)(uintptr_t)gsrc)
               : "memory");
}
__device__ __forceinline__ void async_wait0() {
  asm volatile("s_wait_asynccnt 0x0" ::: "memory");
}

// ---------------------------------------------------------------------------
// Kernel 1: h = LN( ELU( LN(x) @ W1 + b1 ) )    x:[N,128] -> h:[N,64]
// Block = 256 thr (8 waves), 128 rows/block, wave owns 16 rows.
// ---------------------------------------------------------------------------
__global__ __launch_bounds__(256) void fc_first_kernel(
    const float* __restrict__ x,
    const float* __restrict__ ln1_g, const float* __restrict__ ln1_b,
    const float* __restrict__ w1, const float* __restrict__ b1,
    const float* __restrict__ ln_g, const float* __restrict__ ln_b,
    float* __restrict__ h) {
  extern __shared__ float lds[];
  float* Alds = lds;              // [128][132]  LN(x) tile (padded stride)
  float* Wlds = lds + 128 * 132;  // [128][68]   W1 (padded stride)

  const int tid = threadIdx.x;
  const int wave = tid >> 5;
  const int lane = tid & 31;
  const int l15 = lane & 15;
  const int half = lane >> 4;  // which K/M half this lane serves
  const int rowoff = half * 8;
  const int blockRow0 = blockIdx.x * 128;
  const int wr0 = wave * 16;  // wave's row offset within block

  // Stage W1 [128x64] into LDS via async copies (16B granules)
  for (int i = tid; i < FIN * 16; i += 256) {
    int r = i >> 4, c = (i & 15) * 4;
    async_copy16(Wlds + r * 68 + c, w1 + r * 64 + c);
  }

  // Per-row LayerNorm of x into LDS (wave handles its own 16 rows).
  // x is streamed once -> non-temporal so it does not evict the L2-resident
  // h/agg working set used by the scatter phase.
  for (int r = 0; r < 16; ++r) {
    int row = blockRow0 + wr0 + r;
    v4f v = __builtin_nontemporal_load(
        (const v4f*)(x + (size_t)row * FIN + lane * 4));
    float s = wave_sum32(v.x + v.y + v.z + v.w);
    float sq = wave_sum32(v.x * v.x + v.y * v.y + v.z * v.z + v.w * v.w);
    float mean = s * (1.f / FIN);
    float rstd = rsqrtf(sq * (1.f / FIN) - mean * mean + LN_EPS);
    int c = lane * 4;
    v4f o;
    o.x = (v.x - mean) * rstd * ln1_g[c + 0] + ln1_b[c + 0];
    o.y = (v.y - mean) * rstd * ln1_g[c + 1] + ln1_b[c + 1];
    o.z = (v.z - mean) * rstd * ln1_g[c + 2] + ln1_b[c + 2];
    o.w = (v.w - mean) * rstd * ln1_g[c + 3] + ln1_b[c + 3];
    *(v4f*)(Alds + (wr0 + r) * 132 + c) = o;
  }
  async_wait0();
  __syncthreads();

  // GEMM: 16x64 tile per wave via f32 WMMA, K = 128
  v8f acc[4] = {};
  for (int k = 0; k < FIN; k += 4) {
    v2f a = *(const v2f*)(Alds + (wr0 + l15) * 132 + k + 2 * half);
#pragma unroll
    for (int n = 0; n < 4; ++n) {
      const float* bp = Wlds + (k + 2 * half) * 68 + n * 16 + l15;
      v2f b;
      b.x = bp[0];
      b.y = bp[68];
      acc[n] = wmma4(a, b, acc[n]);
    }
  }

  // Epilogue: +b1, ELU, stash rows back to LDS for the second LayerNorm
#pragma unroll
  for (int n = 0; n < 4; ++n) {
    int col = n * 16 + l15;
    float bb = b1[col];
#pragma unroll
    for (int r = 0; r < 8; ++r)
      Alds[(wr0 + rowoff + r) * 132 + col] = elu1(acc[n][r] + bb);
  }

  // LN over H=64 and store h
  for (int r = 0; r < 16; ++r) {
    const float* zr = Alds + (wr0 + r) * 132;
    float z0 = zr[lane * 2], z1 = zr[lane * 2 + 1];
    float s = wave_sum32(z0 + z1);
    float sq = wave_sum32(z0 * z0 + z1 * z1);
    float mean = s * (1.f / HD);
    float rstd = rsqrtf(sq * (1.f / HD) - mean * mean + LN_EPS);
    int c0 = lane * 2;
    int row = blockRow0 + wr0 + r;
    h[(size_t)row * HD + c0] = (z0 - mean) * rstd * ln_g[c0] + ln_b[c0];
    h[(size_t)row * HD + c0 + 1] =
        (z1 - mean) * rstd * ln_g[c0 + 1] + ln_b[c0 + 1];
  }
}

// ---------------------------------------------------------------------------
// Kernel 2: edge scatter  agg[dst] += h[src] * w   (L2-resident f32 atomics)
// 16 threads per edge, float4 channels each.
// ---------------------------------------------------------------------------
__global__ __launch_bounds__(256) void scatter_kernel(
    const float* __restrict__ h, const float* __restrict__ ew,
    const int* __restrict__ src, const int* __restrict__ dst,
    float* __restrict__ agg) {
  int id = blockIdx.x * 256 + threadIdx.x;
  int e = id >> 4;
  int c = (id & 15) * 4;
  int s = src[e], d = dst[e];
  float w = ew[e];
  v4f v = *(const v4f*)(h + (size_t)s * HD + c);
  float* dp = agg + (size_t)d * HD + c;
  atomicAdd(dp + 0, v.x * w);
  atomicAdd(dp + 1, v.y * w);
  atomicAdd(dp + 2, v.z * w);
  atomicAdd(dp + 3, v.w * w);
}

// ---------------------------------------------------------------------------
// Kernel 3: h = LN( agg @ Wc + bc ) + h        (note: res_sum == h identity)
// ---------------------------------------------------------------------------
__global__ __launch_bounds__(256) void conv_kernel(
    const float* __restrict__ agg, const float* __restrict__ wc,
    const float* __restrict__ bc, const float* __restrict__ ln_g,
    const float* __restrict__ ln_b, float* __restrict__ h) {
  extern __shared__ float lds[];
  float* Alds = lds;             // [128][68]
  float* Wlds = lds + 128 * 68;  // [64][68]

  const int tid = threadIdx.x;
  const int wave = tid >> 5;
  const int lane = tid & 31;
  const int l15 = lane & 15;
  const int half = lane >> 4;
  const int rowoff = half * 8;
  const int blockRow0 = blockIdx.x * 128;
  const int wr0 = wave * 16;

  // Async-stage the agg tile [128x64] and Wc [64x64] into LDS
  for (int i = tid; i < 128 * 16; i += 256) {
    int r = i >> 4, c = (i & 15) * 4;
    async_copy16(Alds + r * 68 + c, agg + (size_t)(blockRow0 + r) * HD + c);
  }
  for (int i = tid; i < 64 * 16; i += 256) {
    int r = i >> 4, c = (i & 15) * 4;
    async_copy16(Wlds + r * 68 + c, wc + r * 64 + c);
  }
  async_wait0();
  __syncthreads();

  v8f acc[4] = {};
  for (int k = 0; k < HD; k += 4) {
    v2f a = *(const v2f*)(Alds + (wr0 + l15) * 68 + k + 2 * half);
#pragma unroll
    for (int n = 0; n < 4; ++n) {
      const float* bp = Wlds + (k + 2 * half) * 68 + n * 16 + l15;
      v2f b;
      b.x = bp[0];
      b.y = bp[68];
      acc[n] = wmma4(a, b, acc[n]);
    }
  }

#pragma unroll
  for (int n = 0; n < 4; ++n) {
    int col = n * 16 + l15;
    float bb = bc[col];
#pragma unroll
    for (int r = 0; r < 8; ++r)
      Alds[(wr0 + rowoff + r) * 68 + col] = acc[n][r] + bb;
  }

  for (int r = 0; r < 16; ++r) {
    const float* zr = Alds + (wr0 + r) * 68;
    float z0 = zr[lane * 2], z1 = zr[lane * 2 + 1];
    float s = wave_sum32(z0 + z1);
    float sq = wave_sum32(z0 * z0 + z1 * z1);
    float mean = s * (1.f / HD);
    float rstd = rsqrtf(sq * (1.f / HD) - mean * mean + LN_EPS);
    int c0 = lane * 2;
    int row = blockRow0 + wr0 + r;
    float hn0 = (z0 - mean) * rstd * ln_g[c0] + ln_b[c0];
    float hn1 = (z1 - mean) * rstd * ln_g[c0 + 1] + ln_b[c0 + 1];
    float* hp = h + (size_t)row * HD + c0;
    hp[0] = hp[0] + hn0;  // residual: h_new = LN(...) + h_old (== res_sum)
    hp[1] = hp[1] + hn1;
  }
}

// ---------------------------------------------------------------------------
// Kernel 4: out = ELU( LN(h) @ W3 + b3 ) @ W4 + b4   -> [N,32]
// Two chained WMMA GEMMs per wave (z kept in LDS).
// ---------------------------------------------------------------------------
__global__ __launch_bounds__(256) void fc_final_kernel(
    const float* __restrict__ h, const float* __restrict__ ln2_g,
    const float* __restrict__ ln2_b, const float* __restrict__ w3,
    const float* __restrict__ b3, const float* __restrict__ w4,
    const float* __restrict__ b4, float* __restrict__ out) {
  extern __shared__ float lds[];
  float* Alds = lds;               // [128][68]
  float* W3lds = lds + 128 * 68;   // [64][68]
  float* W4lds = W3lds + 64 * 68;  // [64][36]

  const int tid = threadIdx.x;
  const int wave = tid >> 5;
  const int lane = tid & 31;
  const int l15 = lane & 15;
  const int half = lane >> 4;
  const int rowoff = half * 8;
  const int blockRow0 = blockIdx.x * 128;
  const int wr0 = wave * 16;

  // Async-stage W3 [64x64] and W4 [64x32]
  for (int i = tid; i < 64 * 16; i += 256) {
    int r = i >> 4, c = (i & 15) * 4;
    async_copy16(W3lds + r * 68 + c, w3 + r * 64 + c);
  }
  for (int i = tid; i < 64 * 8; i += 256) {
    int r = i >> 3, c = (i & 7) * 4;
    async_copy16(W4lds + r * 36 + c, w4 + r * 32 + c);
  }

  for (int r = 0; r < 16; ++r) {
    int row = blockRow0 + wr0 + r;
    float2 v = *(const float2*)(h + (size_t)row * HD + lane * 2);
    float s = wave_sum32(v.x + v.y);
    float sq = wave_sum32(v.x * v.x + v.y * v.y);
    float mean = s * (1.f / HD);
    float rstd = rsqrtf(sq * (1.f / HD) - mean * mean + LN_EPS);
    int c0 = lane * 2;
    float* ar = Alds + (wr0 + r) * 68;
    ar[c0] = (v.x - mean) * rstd * ln2_g[c0] + ln2_b[c0];
    ar[c0 + 1] = (v.y - mean) * rstd * ln2_g[c0 + 1] + ln2_b[c0 + 1];
  }
  async_wait0();
  __syncthreads();

  // GEMM1: z = ELU(A @ W3 + b3)
  v8f acc[4] = {};
  for (int k = 0; k < HD; k += 4) {
    v2f a = *(const v2f*)(Alds + (wr0 + l15) * 68 + k + 2 * half);
#pragma unroll
    for (int n = 0; n < 4; ++n) {
      const float* bp = W3lds + (k + 2 * half) * 68 + n * 16 + l15;
      v2f b;
      b.x = bp[0];
      b.y = bp[68];
      acc[n] = wmma4(a, b, acc[n]);
    }
  }
#pragma unroll
  for (int n = 0; n < 4; ++n) {
    int col = n * 16 + l15;
    float bb = b3[col];
#pragma unroll
    for (int r = 0; r < 8; ++r)
      Alds[(wr0 + rowoff + r) * 68 + col] = elu1(acc[n][r] + bb);
  }

  // GEMM2: out = z @ W4 + b4
  v8f acc2[2] = {};
  for (int k = 0; k < HD; k += 4) {
    v2f a = *(const v2f*)(Alds + (wr0 + l15) * 68 + k + 2 * half);
#pragma unroll
    for (int n = 0; n < 2; ++n) {
      const float* bp = W4lds + (k + 2 * half) * 36 + n * 16 + l15;
      v2f b;
      b.x = bp[0];
      b.y = bp[36];
      acc2[n] = wmma4(a, b, acc2[n]);
    }
  }
#pragma unroll
  for (int n = 0; n < 2; ++n) {
    int col = n * 16 + l15;
    float bb = b4[col];
#pragma unroll
    for (int r = 0; r < 8; ++r) {
      int row = blockRow0 + wr0 + rowoff + r;
      __builtin_nontemporal_store(acc2[n][r] + bb,
                                  out + (size_t)row * OD + col);
    }
  }
}

// ---------------------------------------------------------------------------
extern "C" void kernel_launch(void* const* d_in, const int* in_sizes, int n_in,
                              void* d_out, int out_size, void* d_ws,
                              size_t ws_size, hipStream_t stream) {
  (void)in_sizes;
  (void)n_in;
  (void)out_size;
  (void)ws_size;
  const float* x = (const float*)d_in[0];
  const float* ew = (const float*)d_in[1];
  const int* src = (const int*)d_in[2];
  const int* dst = (const int*)d_in[3];
  const float* ln1_g = (const float*)d_in[4];
  const float* ln1_b = (const float*)d_in[5];
  const float* w1 = (const float*)d_in[6];
  const float* b1 = (const float*)d_in[7];
  const float* ln_g = (const float*)d_in[8];
  const float* ln_b = (const float*)d_in[9];
  const float* conv_w = (const float*)d_in[10];
  const float* conv_b = (const float*)d_in[11];
  const float* ln2_g = (const float*)d_in[12];
  const float* ln2_b = (const float*)d_in[13];
  const float* w3 = (const float*)d_in[14];
  const float* b3 = (const float*)d_in[15];
  const float* w4 = (const float*)d_in[16];
  const float* b4 = (const float*)d_in[17];
  float* out = (float*)d_out;

  float* h_buf = (float*)d_ws;           // 16 MB
  float* agg = h_buf + (size_t)NN * HD;  // 16 MB

  const size_t lds1 = (size_t)(128 * 132 + 128 * 68) * sizeof(float);
  const size_t lds2 = (size_t)(128 * 68 + 64 * 68) * sizeof(float);
  const size_t lds3 = (size_t)(128 * 68 + 64 * 68 + 64 * 36) * sizeof(float);

  fc_first_kernel<<<NN / 128, 256, lds1, stream>>>(x, ln1_g, ln1_b, w1, b1,
                                                   ln_g, ln_b, h_buf);
  for (int l = 0; l < NL; ++l) {
    (void)hipMemsetAsync(agg, 0, (size_t)NN * HD * sizeof(float), stream);
    scatter_kernel<<<(NE * 16) / 256, 256, 0, stream>>>(h_buf, ew, src, dst,
                                                        agg);
    conv_kernel<<<NN / 128, 256, lds2, stream>>>(
        agg, conv_w + (size_t)l * HD * HD, conv_b + (size_t)l * HD, ln_g, ln_b,
        h_buf);
  }
  fc_final_kernel<<<NN / 128, 256, lds3, stream>>>(h_buf, ln2_g, ln2_b, w3, b3,
                                                   w4, b4, out);
}